// GNN_conv_28836410425908
// MI455X (gfx1250) — compile-verified
//
#include <hip/hip_runtime.h>

#define N_NODES 100000
#define N_EDGES 1600000
#define IN_FT   128
#define OUT_FT  128

typedef __attribute__((ext_vector_type(2))) float v2f;
typedef __attribute__((ext_vector_type(8))) float v8f;

// ---------------------------------------------------------------------------
// Kernel 1: h = x @ W + b   via V_WMMA_F32_16X16X4_F32 (fp32 in, fp32 out)
// One wave per 16-row M-tile; 8 N-tiles (full 128 cols) accumulated in regs.
// W staged in LDS (64 KB). All 32 A-fragments preloaded up front so global
// latency is hidden behind the WMMA/LDS stream (no per-iteration loadcnt stall).
// ---------------------------------------------------------------------------
__global__ __launch_bounds__(256) void gemm_bias_wmma(
    const float* __restrict__ x, const float* __restrict__ W,
    const float* __restrict__ b, float* __restrict__ h, int nNodes)
{
    __shared__ float Ws[IN_FT * OUT_FT];   // 64 KB

    // Cooperative load of W into LDS: 256 threads * 16 float4 each.
    {
        const float4* src = (const float4*)W;
        float4*       dst = (float4*)Ws;
        int tid = threadIdx.x;
#pragma unroll
        for (int i = 0; i < 16; ++i)
            dst[tid + i * 256] = src[tid + i * 256];
    }
    __syncthreads();

    const int wave  = threadIdx.x >> 5;
    const int lane  = threadIdx.x & 31;
    const int tile  = blockIdx.x * 8 + wave;         // M-tile index
    if (tile * 16 >= nNodes) return;                 // whole-wave guard (EXEC all-1 inside)

    const int row0  = tile * 16;
    const int m     = lane & 15;                     // M (A) or N-within-tile (B/C)
    const int khalf = lane >> 4;                     // 0: K pair {0,1}; 1: K pair {2,3}

    // ---- Preload all A fragments (issue 32 global_load_b64 early) ----------
    // Lane layout (ISA 7.12.2, fp32 A 16x4): lane m holds x[row0+m][k+2*khalf+{0,1}]
    const v2f* xv = (const v2f*)(x + (size_t)(row0 + m) * IN_FT + 2 * khalf);
    v2f afrag[32];
#pragma unroll
    for (int kk = 0; kk < 32; ++kk)
        afrag[kk] = xv[kk * 2];                      // float offset kk*4

    v8f acc[8] = {};                                 // 8 N-tiles of 16x16 f32

#pragma unroll
    for (int kk = 0; kk < 32; ++kk) {
        const int krow = kk * 4 + 2 * khalf;
#pragma unroll
        for (int nt = 0; nt < 8; ++nt) {
            const int n = nt * 16 + m;
            v2f bf;
            bf.x = Ws[(krow + 0) * OUT_FT + n];
            bf.y = Ws[(krow + 1) * OUT_FT + n];
            acc[nt] = __builtin_amdgcn_wmma_f32_16x16x4_f32(
                /*neg_a=*/false, afrag[kk], /*neg_b=*/false, bf,
                /*c_mod=*/(short)0, acc[nt],
                /*reuse_a=*/false, /*reuse_b=*/false);
        }
    }

    // Epilogue: C/D layout — VGPR j: lanes 0-15 -> M=j, lanes 16-31 -> M=j+8.
#pragma unroll
    for (int nt = 0; nt < 8; ++nt) {
        const int n      = nt * 16 + m;
        const float bias = b[n];
#pragma unroll
        for (int j = 0; j < 8; ++j) {
            const int row = row0 + j + 8 * khalf;
            h[(size_t)row * OUT_FT + n] = acc[nt][j] + bias;
        }
    }
}

// ---------------------------------------------------------------------------
// Kernel 0: zero the output accumulator (d_out is poisoned by the harness).
// ---------------------------------------------------------------------------
__global__ __launch_bounds__(256) void zero_f32x4(float4* __restrict__ p, int n4)
{
    int i = blockIdx.x * blockDim.x + threadIdx.x;
    if (i < n4) p[i] = make_float4(0.f, 0.f, 0.f, 0.f);
}

// ---------------------------------------------------------------------------
// Kernel 2: edge scatter.  One wave per edge: 32 lanes x float4 = 512 B row.
// Edge index forced into an SGPR so rows/cols/vals come in via s_load (KMcnt),
// leaving the vector pipe for the h gather (L2-resident) and the f32 atomics.
// ---------------------------------------------------------------------------
__global__ __launch_bounds__(256) void spmm_scatter(
    const float* __restrict__ h,
    const int*   __restrict__ rows,
    const int*   __restrict__ cols,
    const float* __restrict__ vals,
    float* __restrict__ out, int nEdges)
{
    // wave-uniform edge index, pinned to an SGPR
    const int gwave = __builtin_amdgcn_readfirstlane(
        blockIdx.x * 8 + (threadIdx.x >> 5));
    const int lane  = threadIdx.x & 31;
    if (gwave >= nEdges) return;

    const int   r = rows[gwave];     // scalar loads (uniform index)
    const int   c = cols[gwave];
    const float v = vals[gwave];

    const float4 mv = *((const float4*)(h + (size_t)c * OUT_FT) + lane);

    float* op = out + (size_t)r * OUT_FT + lane * 4;
    unsafeAtomicAdd(op + 0, v * mv.x);   // global_atomic_add_f32 (no return)
    unsafeAtomicAdd(op + 1, v * mv.y);
    unsafeAtomicAdd(op + 2, v * mv.z);
    unsafeAtomicAdd(op + 3, v * mv.w);
}

// ---------------------------------------------------------------------------
// Kernel 3: in-place ReLU on out.
// ---------------------------------------------------------------------------
__global__ __launch_bounds__(256) void relu_f32x4(float4* __restrict__ p, int n4)
{
    int i = blockIdx.x * blockDim.x + threadIdx.x;
    if (i < n4) {
        float4 t = p[i];
        t.x = fmaxf(t.x, 0.f);
        t.y = fmaxf(t.y, 0.f);
        t.z = fmaxf(t.z, 0.f);
        t.w = fmaxf(t.w, 0.f);
        p[i] = t;
    }
}

extern "C" void kernel_launch(void* const* d_in, const int* in_sizes, int n_in,
                              void* d_out, int out_size, void* d_ws, size_t ws_size,
                              hipStream_t stream)
{
    const float* x         = (const float*)d_in[0];   // [N_NODES, IN_FT]
    const int*   edge_rows = (const int*)  d_in[1];   // [N_EDGES]
    const int*   edge_cols = (const int*)  d_in[2];   // [N_EDGES]
    const float* edge_vals = (const float*)d_in[3];   // [N_EDGES]
    const float* W         = (const float*)d_in[4];   // [IN_FT, OUT_FT]
    const float* b         = (const float*)d_in[5];   // [OUT_FT]
    float*       out       = (float*)d_out;           // [N_NODES, OUT_FT]
    float*       h         = (float*)d_ws;            // [N_NODES, OUT_FT] scratch (51.2 MB)

    const int nOut4 = (N_NODES * OUT_FT) / 4;         // 3.2M float4

    // 1) zero accumulator
    zero_f32x4<<<(nOut4 + 255) / 256, 256, 0, stream>>>((float4*)out, nOut4);

    // 2) h = x @ W + b  (WMMA fp32). 6250 M-tiles, 8 waves/block -> 782 blocks.
    const int mTiles = N_NODES / 16;                  // 6250 exact
    gemm_bias_wmma<<<(mTiles + 7) / 8, 256, 0, stream>>>(x, W, b, h, N_NODES);

    // 3) edge scatter: 1 wave per edge, 8 waves per block -> 200000 blocks.
    spmm_scatter<<<(N_EDGES + 7) / 8, 256, 0, stream>>>(
        h, edge_rows, edge_cols, edge_vals, out, N_EDGES);

    // 4) ReLU in place
    relu_f32x4<<<(nOut4 + 255) / 256, 256, 0, stream>>>((float4*)out, nOut4);
}